// SparseInst_71116068487974
// MI455X (gfx1250) — compile-verified
//
#include <hip/hip_runtime.h>
#include <math.h>

// ---------------------------------------------------------------------------
// SparseInst post-processing for MI455X (gfx1250, wave32).
// Heavy GEMMs (mn[112x128] @ encode[128x409600]) use V_WMMA_F32_16X16X4_F32
// for full f32 precision (masks pass through a 0.45 threshold).
// GEMM is run twice (stats pass + permuted write pass) to avoid materializing
// the 164MB unsorted mask tensor: ~585MB total HBM traffic @ 23.3 TB/s.
// A-matrix staging (57KB global->LDS per block) uses the CDNA5 Tensor Data
// Mover (tensor_load_to_lds + s_wait_tensorcnt); this toolchain exposes the
// 6-argument builtin (clang-23 / therock-10.0 form).
// ---------------------------------------------------------------------------

typedef __attribute__((ext_vector_type(2))) float v2f;
typedef __attribute__((ext_vector_type(8))) float v8f;
typedef __attribute__((ext_vector_type(4))) unsigned int v4u;
typedef __attribute__((ext_vector_type(8))) int v8i;
typedef __attribute__((ext_vector_type(4))) int v4i;

#define HW    25600      // 160*160
#define DIM   128
#define KCAP  100
#define MPAD  112        // 7 M-tiles of 16
#define NTOT  409600     // 640*640
#define W640  640
#define DSN   102400     // 320*320 downsampled mask
#define DSW   3200       // packed u32 words per ds row

// ---- workspace byte offsets -----------------------------------------------
#define WS_TOPS    0                       // 128 f
#define WS_IDX     512                     // 128 i32
#define WS_VALID   1024                    // 128 f
#define WS_FEAT    1536                    // 100*128 f
#define WS_FN      (WS_FEAT + 51200)       // 100*128 f
#define WS_LABEL   (WS_FN + 51200)         // 100*100 u8 (pad 10240)
#define WS_LABELK  (WS_LABEL + 10240)      // 100*100 u8
#define WS_NORM    (WS_LABELK + 10240)     // 128 f
#define WS_KEEPF   (WS_NORM + 512)         // 128 f
#define WS_AMAT    (WS_KEEPF + 512)        // 112*128 f = 57344
#define WS_SUMS    (WS_AMAT + 57344)       // 128 f
#define WS_CNTS    (WS_SUMS + 512)         // 128 f
#define WS_ORDER   (WS_CNTS + 512)         // 128 i32
#define WS_SCS     (WS_ORDER + 512)        // 128 f
#define WS_KEEPS   (WS_SCS + 512)          // 128 f
#define WS_KEEP2   (WS_KEEPS + 512)        // 128 f
#define WS_INTER   (WS_KEEP2 + 512)        // 100*100 f (pad 40960)
#define WS_A2      (WS_INTER + 40960)      // 112*128 f
#define WS_DS      (WS_A2 + 57344)         // 100*102400 u8 = 10,240,000
#define WS_PACK    (WS_DS + 10240000)      // 100*3200 u32 = 1,280,000
// total ~ 11.8 MB

// ---------------------------------------------------------------------------
// TDM staging of the 112x128 f32 A matrix into LDS (one descriptor per block).
// D# bitfields per CDNA5 ISA ch.8 (groups 0/1; 2D tensor so trailing groups
// are zero). 6-arg builtin form: (v4u g0, v8i g1, v4i, v4i, v8i, i32 cpol).
// ---------------------------------------------------------------------------
#if defined(__gfx1250__) && __has_builtin(__builtin_amdgcn_tensor_load_to_lds)
#define HAVE_TDM 1
#else
#define HAVE_TDM 0
#endif

__device__ __forceinline__ void stage_A_to_lds(const float* __restrict__ gsrc,
                                               float* lds_dst, int t) {
#if HAVE_TDM
  // generic LDS address: low 32 bits are the LDS byte offset (aperture spec)
  unsigned long long ga = (unsigned long long)(uintptr_t)gsrc;
  unsigned ldsOff = (unsigned)(uintptr_t)lds_dst;
  v4u g0;
  g0[0] = 1u;                                       // count=1, user mode
  g0[1] = ldsOff;                                   // lds_addr
  g0[2] = (unsigned)(ga & 0xFFFFFFFFu);             // global_addr[31:0]
  g0[3] = (unsigned)((ga >> 32) & 0x01FFFFFFu)      // global_addr[56:32]
          | (2u << 30);                             // type = 2 ("image")
  v8i g1;
  g1[0] = (int)(2u << 16);                          // wg_mask=0, data_size=4B
  g1[1] = (int)(128u << 16);                        // tensor_dim0=128 (bits 63:48)
  g1[2] = (int)(112u << 16);                        // dim0 hi=0 | tensor_dim1=112
  g1[3] = (int)(128u << 16);                        // dim1 hi=0 | tile_dim0=128
  g1[4] = (int)112u;                                // tile_dim1=112, tile_dim2=0
  g1[5] = (int)128u;                                // tensor_dim0_stride=128
  g1[6] = 0;                                        // stride hi | dim1_stride lo
  g1[7] = 0;
  v4i gz4 = {0, 0, 0, 0};
  v8i gz8 = {0, 0, 0, 0, 0, 0, 0, 0};
  if (t < 32) {                                     // one wave issues the DMA
    __builtin_amdgcn_tensor_load_to_lds(g0, g1, gz4, gz4, gz8, 0);
    __builtin_amdgcn_s_wait_tensorcnt(0);
  }
#else
  for (int i = t; i < MPAD * DIM; i += 256) lds_dst[i] = gsrc[i];
#endif
}

// ---------------------------------------------------------------------------
// 1) sigmoid + top-100 (sigmoid is monotonic -> top-k on raw logits)
// ---------------------------------------------------------------------------
__global__ __launch_bounds__(1024) void k_topk(const float* __restrict__ logits,
                                               float* top_s, int* idxArr, float* validf) {
  __shared__ float sv[1024];
  __shared__ int   si[1024];
  __shared__ unsigned int bm[800];   // 25600-bit "already selected" bitmap
  int t = threadIdx.x;
  for (int i = t; i < 800; i += 1024) bm[i] = 0u;
  __syncthreads();
  for (int sel = 0; sel < KCAP; ++sel) {
    float bv = -3.4e38f; int bi = HW;
    for (int i = t; i < HW; i += 1024) {
      if (bm[i >> 5] & (1u << (i & 31))) continue;
      float v = logits[i];
      if (v > bv || (v == bv && i < bi)) { bv = v; bi = i; }
    }
    sv[t] = bv; si[t] = bi;
    __syncthreads();
    for (int s = 512; s > 0; s >>= 1) {
      if (t < s) {
        float ov = sv[t + s]; int oi = si[t + s];
        if (ov > sv[t] || (ov == sv[t] && oi < si[t])) { sv[t] = ov; si[t] = oi; }
      }
      __syncthreads();
    }
    if (t == 0) {
      int b = si[0];
      bm[b >> 5] |= (1u << (b & 31));
      float s = 1.f / (1.f + __expf(-sv[0]));
      top_s[sel]  = s;
      idxArr[sel] = b;
      validf[sel] = (s > 0.75f) ? 1.f : 0.f;
    }
    __syncthreads();
  }
}

// ---------------------------------------------------------------------------
// 2) gather embedding rows + L2-normalize
// ---------------------------------------------------------------------------
__global__ void k_gather(const float* __restrict__ emb, const int* __restrict__ idxArr,
                         float* feat, float* fn) {
  int kk = blockIdx.x, d = threadIdx.x;      // 100 x 128
  __shared__ float red[DIM];
  int id = idxArr[kk];
  float v = emb[(size_t)d * HW + id];
  feat[kk * DIM + d] = v;
  red[d] = v * v;
  __syncthreads();
  for (int st = 64; st > 0; st >>= 1) { if (d < st) red[d] += red[d + st]; __syncthreads(); }
  float rinv = 1.f / fmaxf(sqrtf(red[0]), 1e-8f);
  fn[kk * DIM + d] = v * rinv;
}

// ---------------------------------------------------------------------------
// 3) label[i][j] = triu(cosine sim) >= 0.75 & pair_valid
// ---------------------------------------------------------------------------
__global__ void k_label(const float* __restrict__ fn, const float* __restrict__ validf,
                        unsigned char* labelb) {
  int i = blockIdx.x, j = threadIdx.x;       // 100 x 128
  if (j >= KCAP) return;
  const float* a = fn + i * DIM;
  const float* b = fn + j * DIM;
  float dot = 0.f;
  for (int d = 0; d < DIM; ++d) dot += a[d] * b[d];
  unsigned char l = (unsigned char)((j >= i) && (dot >= 0.75f) &&
                                    (validf[i] > 0.f) && (validf[j] > 0.f));
  labelb[i * KCAP + j] = l;
}

// ---------------------------------------------------------------------------
// 4) column cumsum dedup -> keep, label_k, row norms
// ---------------------------------------------------------------------------
__global__ void k_dedup(const unsigned char* __restrict__ labelb,
                        const float* __restrict__ validf,
                        unsigned char* labelk, float* normArr, float* keepfArr) {
  __shared__ unsigned char lab[KCAP * KCAP];
  __shared__ float keepL[KCAP];
  int t = threadIdx.x;                        // 128
  for (int p = t; p < KCAP * KCAP; p += 128) lab[p] = labelb[p];
  __syncthreads();
  if (t < KCAP) {                             // thread t owns column t
    int c = 0;
    for (int i = 0; i < KCAP; ++i) {
      int l = lab[i * KCAP + t];
      c += l;
      int cum = (c < 2) ? 1 : 0;
      if (i == t) keepL[t] = (cum && validf[t] > 0.f) ? 1.f : 0.f;
      lab[i * KCAP + t] = (unsigned char)(l & cum);
    }
  }
  __syncthreads();
  if (t < KCAP) {                             // thread t owns row t
    float kv = keepL[t];
    float s = 0.f;
    for (int j = 0; j < KCAP; ++j) {
      unsigned char v = (kv > 0.f) ? lab[t * KCAP + j] : (unsigned char)0;
      labelk[t * KCAP + j] = v;
      s += (float)v;
    }
    normArr[t]  = fmaxf(s, 1.f);
    keepfArr[t] = kv;
  } else if (t < MPAD) {
    keepfArr[t] = 0.f;                        // pad rows never kept
  }
}

// ---------------------------------------------------------------------------
// 5) meta = (label_k @ feat)/norm; A = meta / ||meta||  (padded to 112 rows)
// ---------------------------------------------------------------------------
__global__ void k_meta(const float* __restrict__ feat, const unsigned char* __restrict__ labelk,
                       const float* __restrict__ normArr, float* Amat) {
  int i = blockIdx.x, d = threadIdx.x;        // 112 x 128
  __shared__ float red[DIM];
  if (i >= KCAP) { Amat[i * DIM + d] = 0.f; return; }
  float s = 0.f;
  for (int j = 0; j < KCAP; ++j)
    if (labelk[i * KCAP + j]) s += feat[j * DIM + d];
  float meta = s / normArr[i];
  red[d] = meta * meta;
  __syncthreads();
  for (int st = 64; st > 0; st >>= 1) { if (d < st) red[d] += red[d + st]; __syncthreads(); }
  float rinv = 1.f / fmaxf(sqrtf(red[0]), 1e-8f);
  Amat[i * DIM + d] = meta * rinv;
}

__global__ void k_zero(float* sums, float* cnts) {
  int t = threadIdx.x;
  if (t < 128) { sums[t] = 0.f; cnts[t] = 0.f; }
}

// ---------------------------------------------------------------------------
// 6) WMMA GEMM pass 1: inst = sigmoid(A @ enc) * keep; accumulate masked
//    sums/counts per row and emit 0.5x-downsampled binary masks.
//    One wave = one 16-column N-tile; 7 M-tiles; K=128 in steps of 4.
// ---------------------------------------------------------------------------
__global__ __launch_bounds__(256) void k_gemm_stats(
    const float* __restrict__ enc, const float* __restrict__ Amat,
    const float* __restrict__ keepf, float* sums, float* cnts, unsigned char* ds) {
  __shared__ float As[MPAD * DIM];   // 57344 B, well under 320KB/WGP
  __shared__ float kL[MPAD];
  __shared__ float ssum[MPAD];
  __shared__ float scnt[MPAD];
  int t = threadIdx.x;
  stage_A_to_lds(Amat, As, t);                // TDM DMA
  if (t < MPAD) { kL[t] = keepf[t]; ssum[t] = 0.f; scnt[t] = 0.f; }
  __syncthreads();

  int wave = t >> 5, lane = t & 31;
  int half = lane >> 4, r = lane & 15;       // ISA 16x16x4 f32 layout halves
  int tile = blockIdx.x * 8 + wave;          // 3200 blocks * 8 waves = 25600 tiles
  int n0 = tile * 16;
  int n  = n0 + r;

  const v8f vzero = {0.f, 0.f, 0.f, 0.f, 0.f, 0.f, 0.f, 0.f};
  v8f acc[7];
#pragma unroll
  for (int m = 0; m < 7; ++m) acc[m] = vzero;

  const float* bp = enc + (size_t)(2 * half) * NTOT + n;
  for (int k = 0; k < DIM; k += 4) {
    if (k + 4 < DIM) {                        // stream-ahead: global_prefetch_b8
      __builtin_prefetch(bp + (size_t)(k + 4) * NTOT, 0, 1);
      __builtin_prefetch(bp + (size_t)(k + 5) * NTOT, 0, 1);
    }
    v2f b;
    b.x = bp[(size_t)k * NTOT];               // B[k+2*half  ][n]
    b.y = bp[(size_t)(k + 1) * NTOT];         // B[k+2*half+1][n]
#pragma unroll
    for (int m = 0; m < 7; ++m) {
      const float* ap = &As[(m * 16 + r) * DIM + k + 2 * half];
      v2f a; a.x = ap[0]; a.y = ap[1];        // A[M=r][k+2*half .. +1]
      acc[m] = __builtin_amdgcn_wmma_f32_16x16x4_f32(
          false, a, false, b, (short)0, acc[m], false, false);
    }
  }

  int y = n0 / W640;                          // tile never spans rows (640%16==0)
  int x = n - y * W640;
  bool evenRow = ((y & 1) == 0);
#pragma unroll
  for (int m = 0; m < 7; ++m) {
#pragma unroll
    for (int vr = 0; vr < 8; ++vr) {
      int M = m * 16 + vr + 8 * half;         // C layout: halves hold M and M+8
      float v = kL[M] / (1.f + __expf(-acc[m][vr]));
      float msk = (v > 0.45f) ? 1.f : 0.f;
      if (evenRow && ((x & 1) == 0) && M < KCAP)
        ds[(size_t)M * DSN + (size_t)(y >> 1) * 320 + (x >> 1)] = (unsigned char)(msk != 0.f);
      float sv = v * msk;
      for (int o = 1; o < 16; o <<= 1) {      // reduce over the 16 N-lanes of a half
        sv  += __shfl_xor(sv,  o, 32);
        msk += __shfl_xor(msk, o, 32);
      }
      if (r == 0) { atomicAdd(&ssum[M], sv); atomicAdd(&scnt[M], msk); }
    }
  }
  __syncthreads();
  if (t < MPAD) { atomicAdd(&sums[t], ssum[t]); atomicAdd(&cnts[t], scnt[t]); }
}

// ---------------------------------------------------------------------------
// 7) bit-pack downsampled masks for popcount NMS
// ---------------------------------------------------------------------------
__global__ void k_pack(const unsigned char* __restrict__ ds, unsigned int* packed) {
  int w = blockIdx.x * 256 + threadIdx.x;
  if (w >= KCAP * DSW) return;
  int m = w / DSW, wi = w - m * DSW;
  const unsigned char* p = ds + (size_t)m * DSN + (size_t)wi * 32;
  unsigned int u = 0;
  for (int b = 0; b < 32; ++b) u |= ((unsigned int)(p[b] & 1)) << b;
  packed[w] = u;
}

// ---------------------------------------------------------------------------
// 8) score = top_s * masked-mean; stable descending rank (== argsort(-sc))
// ---------------------------------------------------------------------------
__global__ void k_score_sort(const float* __restrict__ top_s, const float* __restrict__ sums,
                             const float* __restrict__ cnts, const float* __restrict__ keepf,
                             int* order, float* scs, float* keeps) {
  __shared__ float scL[KCAP];
  int t = threadIdx.x;                         // 128
  if (t < KCAP) scL[t] = top_s[t] * (sums[t] / fmaxf(cnts[t], 1.f));
  __syncthreads();
  if (t < KCAP) {
    float mv = scL[t];
    int rnk = 0;
    for (int j = 0; j < KCAP; ++j) {
      float o = scL[j];
      if (o > mv || (o == mv && j < t)) ++rnk;
    }
    order[rnk] = t;
    scs[rnk]   = mv;
    keeps[rnk] = keepf[t];
  }
}

// ---------------------------------------------------------------------------
// 9) pairwise intersections via popcount (rows indexed through `order`)
// ---------------------------------------------------------------------------
__global__ void k_inter(const unsigned int* __restrict__ packed, const int* __restrict__ order,
                        float* inter) {
  int i = blockIdx.x, j = blockIdx.y;
  if (j < i) return;
  const unsigned int* a = packed + (size_t)order[i] * DSW;
  const unsigned int* b = packed + (size_t)order[j] * DSW;
  int t = threadIdx.x;                         // 32 = 1 wave
  int acc = 0;
  for (int w = t; w < DSW; w += 32) acc += __popc(a[w] & b[w]);
  for (int o = 1; o < 32; o <<= 1) acc += __shfl_xor(acc, o, 32);
  if (t == 0) inter[i * KCAP + j] = (float)acc;
}

// ---------------------------------------------------------------------------
// 10) matrix-NMS decay + final scores/labels/keep
// ---------------------------------------------------------------------------
__global__ void k_nms_final(const float* __restrict__ inter, const float* __restrict__ scs,
                            const float* __restrict__ keeps, float* keep2f,
                            float* outScores, float* outLabels, float* outKeep) {
  __shared__ float s2[KCAP];
  __shared__ float iouL[KCAP * KCAP];          // 40 KB
  __shared__ float compL[KCAP];
  int t = threadIdx.x;                          // 128
  if (t < KCAP) s2[t] = inter[t * KCAP + t];
  __syncthreads();
  for (int p = t; p < KCAP * KCAP; p += 128) {
    int i = p / KCAP, j = p - i * KCAP;
    float v = 0.f;
    if (i < j) {
      float in = inter[i * KCAP + j];
      float un = fmaxf(s2[i] + s2[j] - in, 1e-6f);
      v = in / un;
    }
    iouL[p] = v;
  }
  __syncthreads();
  if (t < KCAP) {
    float mx = 0.f;
    for (int i = 0; i < KCAP; ++i) mx = fmaxf(mx, iouL[i * KCAP + t]);
    compL[t] = mx;
  }
  __syncthreads();
  if (t < KCAP) {
    float cf = 3.4e38f;
    for (int i = 0; i < KCAP; ++i) {
      float u = iouL[i * KCAP + t];
      float d  = __expf(-2.f * u * u);
      float c  = compL[i];
      float cm = __expf(-2.f * c * c);
      cf = fminf(cf, d / cm);
    }
    float sc = scs[t] * cf;
    float k2 = (sc >= 0.05f && keeps[t] > 0.f) ? 1.f : 0.f;
    outScores[t] = sc * k2;
    outLabels[t] = 0.f;
    outKeep[t]   = k2;
    keep2f[t]    = k2;
  }
  if (t >= KCAP && t < MPAD) keep2f[t] = 0.f;
}

// ---------------------------------------------------------------------------
// 11) build permuted A2 rows for the output GEMM
// ---------------------------------------------------------------------------
__global__ void k_permA(const float* __restrict__ Amat, const int* __restrict__ order,
                        float* A2) {
  int i = blockIdx.x, d = threadIdx.x;          // 112 x 128
  A2[i * DIM + d] = (i < KCAP) ? Amat[order[i] * DIM + d] : 0.f;
}

// ---------------------------------------------------------------------------
// 12) WMMA GEMM pass 2: masks[i] = sigmoid(A2[i] @ enc) * keep2f[i] -> d_out
// ---------------------------------------------------------------------------
__global__ __launch_bounds__(256) void k_gemm_write(
    const float* __restrict__ enc, const float* __restrict__ A2,
    const float* __restrict__ keep2f, float* __restrict__ out) {
  __shared__ float As[MPAD * DIM];
  __shared__ float kL[MPAD];
  int t = threadIdx.x;
  stage_A_to_lds(A2, As, t);                  // TDM DMA
  if (t < MPAD) kL[t] = keep2f[t];
  __syncthreads();

  int wave = t >> 5, lane = t & 31;
  int half = lane >> 4, r = lane & 15;
  int tile = blockIdx.x * 8 + wave;
  int n0 = tile * 16;
  int n  = n0 + r;

  const v8f vzero = {0.f, 0.f, 0.f, 0.f, 0.f, 0.f, 0.f, 0.f};
  v8f acc[7];
#pragma unroll
  for (int m = 0; m < 7; ++m) acc[m] = vzero;

  const float* bp = enc + (size_t)(2 * half) * NTOT + n;
  for (int k = 0; k < DIM; k += 4) {
    if (k + 4 < DIM) {
      __builtin_prefetch(bp + (size_t)(k + 4) * NTOT, 0, 1);
      __builtin_prefetch(bp + (size_t)(k + 5) * NTOT, 0, 1);
    }
    v2f b;
    b.x = bp[(size_t)k * NTOT];
    b.y = bp[(size_t)(k + 1) * NTOT];
#pragma unroll
    for (int m = 0; m < 7; ++m) {
      const float* ap = &As[(m * 16 + r) * DIM + k + 2 * half];
      v2f a; a.x = ap[0]; a.y = ap[1];
      acc[m] = __builtin_amdgcn_wmma_f32_16x16x4_f32(
          false, a, false, b, (short)0, acc[m], false, false);
    }
  }

#pragma unroll
  for (int m = 0; m < 7; ++m) {
#pragma unroll
    for (int vr = 0; vr < 8; ++vr) {
      int M = m * 16 + vr + 8 * half;
      if (M < KCAP) {
        float v = kL[M] / (1.f + __expf(-acc[m][vr]));
        out[(size_t)M * NTOT + n] = v;
      }
    }
  }
}

// ---------------------------------------------------------------------------
extern "C" void kernel_launch(void* const* d_in, const int* in_sizes, int n_in,
                              void* d_out, int out_size, void* d_ws, size_t ws_size,
                              hipStream_t stream) {
  const float* logits = (const float*)d_in[0];   // (1,1,160,160)
  const float* emb    = (const float*)d_in[1];   // (1,128,160,160)
  const float* enc    = (const float*)d_in[2];   // (1,128,640,640)

  char* ws = (char*)d_ws;
  float*         top_s   = (float*)(ws + WS_TOPS);
  int*           idxArr  = (int*)  (ws + WS_IDX);
  float*         validf  = (float*)(ws + WS_VALID);
  float*         feat    = (float*)(ws + WS_FEAT);
  float*         fn      = (float*)(ws + WS_FN);
  unsigned char* labelb  = (unsigned char*)(ws + WS_LABEL);
  unsigned char* labelk  = (unsigned char*)(ws + WS_LABELK);
  float*         normArr = (float*)(ws + WS_NORM);
  float*         keepf   = (float*)(ws + WS_KEEPF);
  float*         Amat    = (float*)(ws + WS_AMAT);
  float*         sums    = (float*)(ws + WS_SUMS);
  float*         cnts    = (float*)(ws + WS_CNTS);
  int*           order   = (int*)  (ws + WS_ORDER);
  float*         scs     = (float*)(ws + WS_SCS);
  float*         keeps   = (float*)(ws + WS_KEEPS);
  float*         keep2f  = (float*)(ws + WS_KEEP2);
  float*         inter   = (float*)(ws + WS_INTER);
  float*         A2      = (float*)(ws + WS_A2);
  unsigned char* ds      = (unsigned char*)(ws + WS_DS);
  unsigned int*  packed  = (unsigned int*)(ws + WS_PACK);

  float* outMasks  = (float*)d_out;                 // (100,640,640)
  float* outScores = outMasks + (size_t)KCAP * NTOT;
  float* outLabels = outScores + KCAP;
  float* outKeep   = outLabels + KCAP;

  k_topk<<<1, 1024, 0, stream>>>(logits, top_s, idxArr, validf);
  k_gather<<<KCAP, DIM, 0, stream>>>(emb, idxArr, feat, fn);
  k_label<<<KCAP, DIM, 0, stream>>>(fn, validf, labelb);
  k_dedup<<<1, DIM, 0, stream>>>(labelb, validf, labelk, normArr, keepf);
  k_meta<<<MPAD, DIM, 0, stream>>>(feat, labelk, normArr, Amat);
  k_zero<<<1, 256, 0, stream>>>(sums, cnts);
  k_gemm_stats<<<3200, 256, 0, stream>>>(enc, Amat, keepf, sums, cnts, ds);
  k_pack<<<(KCAP * DSW + 255) / 256, 256, 0, stream>>>(ds, packed);
  k_score_sort<<<1, DIM, 0, stream>>>(top_s, sums, cnts, keepf, order, scs, keeps);
  k_inter<<<dim3(KCAP, KCAP), 32, 0, stream>>>(packed, order, inter);
  k_nms_final<<<1, DIM, 0, stream>>>(inter, scs, keeps, keep2f, outScores, outLabels, outKeep);
  k_permA<<<MPAD, DIM, 0, stream>>>(Amat, order, A2);
  k_gemm_write<<<3200, 256, 0, stream>>>(enc, A2, keep2f, outMasks);
}